// PinnWithGRU_4329327034799
// MI455X (gfx1250) — compile-verified
//
#include <hip/hip_runtime.h>

typedef __attribute__((ext_vector_type(2))) float v2f;
typedef __attribute__((ext_vector_type(8))) float v8f;

#define B_SZ 2048
#define T_SZ 2048
#define D_SZ 3
#define H_SZ 32

__device__ __forceinline__ float fast_exp(float x) {
    return __builtin_amdgcn_exp2f(x * 1.4426950408889634f);
}
__device__ __forceinline__ float fast_sigmoid(float x) {
    // 1/(1+e^-x); saturates correctly: x->+inf => 1, x->-inf => 0
    return __builtin_amdgcn_rcpf(1.0f + fast_exp(-x));
}
__device__ __forceinline__ float fast_tanh(float x) {
    // 1 - 2/(e^{2x}+1); x->+inf => 1, x->-inf => -1
    return 1.0f - 2.0f * __builtin_amdgcn_rcpf(fast_exp(2.0f * x) + 1.0f);
}

// One wave (32 lanes) owns a tile of 16 batch rows for the whole T recurrence.
// Per step: gates = [x(16x4)|h(16x32)] x [W_ih^T|W_hh^T] via v_wmma_f32_16x16x4_f32,
// 6 N-tiles of 16 gate columns each (96 = 3 gates * H=32 columns).
__global__ __launch_bounds__(32) void gru_wmma_f32_kernel(
    const float* __restrict__ inp,       // [B, T, D]
    const float* __restrict__ w_ih,      // [96, 3]
    const float* __restrict__ w_hh,      // [96, 32]
    const float* __restrict__ bias,      // [96]
    const float* __restrict__ bias_n,    // [32]
    float* __restrict__ out)             // [B, T]
{
    __shared__ float hbuf[16 * 33];      // transpose staging, padded stride

    const int lane = threadIdx.x;        // 0..31
    const int g    = lane >> 4;          // lane group (0: lanes 0-15, 1: 16-31)
    const int ln   = lane & 15;
    const int b0   = blockIdx.x * 16;    // first batch row of this tile

    // ---- W_hh B-fragments: B[k][n] = W_hh[n][k]; frag value v at lane(g,ln):
    //      k = kc*4 + 2*g + v, n = nt*16 + ln. Held in VGPRs for all T steps.
    v2f whh[6][8];
    #pragma unroll
    for (int nt = 0; nt < 6; ++nt) {
        #pragma unroll
        for (int kc = 0; kc < 8; ++kc) {
            const int n = nt * 16 + ln;
            const int k = kc * 4 + 2 * g;
            v2f w = { w_hh[n * H_SZ + k], w_hh[n * H_SZ + k + 1] };
            whh[nt][kc] = w;
        }
    }
    // ---- W_ih B-fragments (K padded 3 -> 4 with zeros)
    v2f wih[6];
    #pragma unroll
    for (int nt = 0; nt < 6; ++nt) {
        const int n = nt * 16 + ln;
        const int k = 2 * g;             // 0 or 2
        float a0 = w_ih[n * D_SZ + k];
        float a1 = (k + 1 < D_SZ) ? w_ih[n * D_SZ + k + 1] : 0.0f;
        v2f w = { a0, a1 };
        wih[nt] = w;
    }
    // ---- per-lane biases for C-layout column j = tl*16 + ln
    float bR[2], bZ[2], bA[2], bN[2];
    #pragma unroll
    for (int tl = 0; tl < 2; ++tl) {
        const int j = tl * 16 + ln;
        bR[tl] = bias[j];
        bZ[tl] = bias[H_SZ + j];
        bA[tl] = bias[2 * H_SZ + j];
        bN[tl] = bias_n[j];
    }

    // ---- state: h in A-layout (feeds WMMA) and C-layout (for z*h blend)
    v2f hA[8];
    #pragma unroll
    for (int kc = 0; kc < 8; ++kc) { v2f z = { 0.0f, 0.0f }; hA[kc] = z; }
    v8f hC[2];
    hC[0] = {}; hC[1] = {};

    // Branchless per-lane x addressing (A layout: both lane groups use row m'=ln):
    //   elem0 offset within row: 3t + 2g   (g0 -> x[0], g1 -> x[2])
    //   elem1 offset within row: 3t + 1+g  (g0 -> x[1], g1 -> x[2], masked to 0)
    const float* xrow = inp + (size_t)(b0 + ln) * T_SZ * D_SZ;
    const float* xp0  = xrow + 2 * g;
    const float* xp1  = xrow + 1 + g;

    // output pointer for the two emitting lanes (ln==0): rows m = 8*g .. 8*g+7
    float* orow = out + (size_t)(b0 + 8 * g) * T_SZ;

    for (int t = 0; t < T_SZ; ++t) {
        // ---- x A-fragment (16x4, K padded), no divergent branches
        float x0 = xp0[3 * t];
        float x1 = xp1[3 * t];
        x1 = g ? 0.0f : x1;              // v_cndmask: pad K=3 lane-half to zero
        v2f xf = { x0, x1 };

        // ---- matmuls on the matrix units
        v8f C[4];                        // r,z gates: x-part + h-part fused
        #pragma unroll
        for (int nt = 0; nt < 4; ++nt) {
            v8f acc = {};
            acc = __builtin_amdgcn_wmma_f32_16x16x4_f32(
                      false, xf, false, wih[nt], (short)0, acc, false, false);
            #pragma unroll
            for (int kc = 0; kc < 8; ++kc)
                acc = __builtin_amdgcn_wmma_f32_16x16x4_f32(
                          false, hA[kc], false, whh[nt][kc], (short)0, acc, false, false);
            C[nt] = acc;
        }
        v8f Cia[2], Cha[2];              // a gate: ia and ha kept separate (r * ha)
        #pragma unroll
        for (int tl = 0; tl < 2; ++tl) {
            v8f accA = {};
            accA = __builtin_amdgcn_wmma_f32_16x16x4_f32(
                       false, xf, false, wih[4 + tl], (short)0, accA, false, false);
            Cia[tl] = accA;
            v8f accH = {};
            #pragma unroll
            for (int kc = 0; kc < 8; ++kc)
                accH = __builtin_amdgcn_wmma_f32_16x16x4_f32(
                           false, hA[kc], false, whh[4 + tl][kc], (short)0, accH, false, false);
            Cha[tl] = accH;
        }

        // ---- gate math (C layout: VGPR r -> row m = r + 8*g, lane ln -> col j)
        #pragma unroll
        for (int tl = 0; tl < 2; ++tl) {
            #pragma unroll
            for (int r = 0; r < 8; ++r) {
                float rg = fast_sigmoid(C[tl][r]     + bR[tl]);
                float zg = fast_sigmoid(C[tl + 2][r] + bZ[tl]);
                float ng = fast_tanh(Cia[tl][r] + bA[tl] + bN[tl] + rg * Cha[tl][r]);
                float hn = ng + zg * (hC[tl][r] - ng);   // (1-z)*n + z*h
                hC[tl][r] = hn;
            }
        }

        // ---- emit h_new[:,0]: column j==0 lives in tile 0 at ln==0 (lanes 0,16)
        if (ln == 0) {
            #pragma unroll
            for (int r = 0; r < 8; ++r)
                orow[(size_t)r * T_SZ + t] = hC[0][r];
        }

        // ---- C-layout -> A-layout transpose through LDS (single-wave WG)
        __syncthreads();
        #pragma unroll
        for (int tl = 0; tl < 2; ++tl) {
            #pragma unroll
            for (int r = 0; r < 8; ++r) {
                const int m = r + 8 * g;
                const int j = tl * 16 + ln;
                hbuf[m * 33 + j] = hC[tl][r];
            }
        }
        __syncthreads();
        #pragma unroll
        for (int kc = 0; kc < 8; ++kc) {
            const int k = kc * 4 + 2 * g;
            v2f hv = { hbuf[ln * 33 + k], hbuf[ln * 33 + k + 1] };
            hA[kc] = hv;
        }
        __syncthreads();
    }
}

extern "C" void kernel_launch(void* const* d_in, const int* in_sizes, int n_in,
                              void* d_out, int out_size, void* d_ws, size_t ws_size,
                              hipStream_t stream) {
    (void)in_sizes; (void)n_in; (void)d_ws; (void)ws_size; (void)out_size;
    const float* inp    = (const float*)d_in[0];   // [B,T,D]
    const float* w_ih   = (const float*)d_in[1];   // [96,3]
    const float* w_hh   = (const float*)d_in[2];   // [96,32]
    const float* bias   = (const float*)d_in[3];   // [96]
    const float* bias_n = (const float*)d_in[4];   // [32]
    float* out = (float*)d_out;                    // [B,T,1]

    dim3 grid(B_SZ / 16);   // 128 tiles of 16 sequences
    dim3 block(32);         // one wave32 per tile
    gru_wmma_f32_kernel<<<grid, block, 0, stream>>>(inp, w_ih, w_hh, bias, bias_n, out);
}